// EdgeDrop_73220602462459
// MI455X (gfx1250) — compile-verified
//
#include <hip/hip_runtime.h>

#define N_NODES 100000
#define N_EDGES 500000
#define KDIM    128     // all layer input dims are 128
#define HDIM    128
#define NCLS    40

typedef __attribute__((ext_vector_type(2))) float v2f;
typedef __attribute__((ext_vector_type(8))) float v8f;

// ---------------- gcn_norm precompute ----------------

__global__ void init_deg_kernel(float* __restrict__ deg, int n) {
    int i = blockIdx.x * blockDim.x + threadIdx.x;
    if (i < n) deg[i] = 1.0f;                    // self-loop contributes 1
}

__global__ void edge_deg_kernel(const int* __restrict__ dst, float* __restrict__ deg, int e) {
    int i = blockIdx.x * blockDim.x + threadIdx.x;
    if (i < e) atomicAdd(&deg[dst[i]], 1.0f);
}

__global__ void dinv_kernel(float* __restrict__ d, int n) {
    int i = blockIdx.x * blockDim.x + threadIdx.x;
    if (i < n) {
        float v = d[i];
        d[i] = (v > 0.0f) ? rsqrtf(v) : 0.0f;    // deg >= 1 always here
    }
}

__global__ void edge_norm_kernel(const int* __restrict__ src, const int* __restrict__ dst,
                                 const float* __restrict__ dinv, float* __restrict__ norm, int e) {
    int i = blockIdx.x * blockDim.x + threadIdx.x;
    if (i < e) norm[i] = dinv[src[i]] * dinv[dst[i]];
}

// ---------------- dense GEMM via V_WMMA_F32_16X16X4_F32 ----------------
// Out[N x FOUT] = act(A)[N x 128] * B[128 x FOUT]
// B staged in LDS once per block, k-pair interleaved:
//   lds[((k>>1)*FP + col)*2 + (k&1)] = B[k][col]
// so each WMMA B fragment (k, k+1 for one column) is one aligned ds_load_b64.
// One wave owns a 16-row strip and all NT = FP/16 column tiles: each k-step
// loads A once (global_load_b64) and issues NT chained WMMAs from LDS.
template <int FOUT, bool RELU>
__global__ __launch_bounds__(256)
void gcn_gemm_wmma_kernel(const float* __restrict__ A,
                          const float* __restrict__ B,
                          float* __restrict__ Out,
                          int totalStrips)
{
    constexpr int FP = (FOUT % 16 == 0) ? FOUT : ((FOUT + 15) & ~15); // padded cols
    constexpr int NT = FP / 16;                                      // 16-col tiles

    __shared__ float ldsB[KDIM * FP];

    // ---- cooperative LDS stage of B (all 256 threads, before any guard) ----
    {
        constexpr int NE = KDIM * FP;
#pragma unroll 4
        for (int p = threadIdx.x; p < NE; p += 256) {
            int k = p / FP;
            int c = p - k * FP;
            float val;
            if constexpr (FOUT == FP)
                val = B[p];                          // contiguous, coalesced
            else
                val = (c < FOUT) ? B[k * FOUT + c] : 0.0f;  // zero-pad cols
            ldsB[(((k >> 1) * FP) + c) * 2 + (k & 1)] = val;
        }
    }
    __syncthreads();

    int strip = blockIdx.x * (blockDim.x >> 5) + (threadIdx.x >> 5);
    if (strip >= totalStrips) return;            // wave-uniform after barrier

    int lane = threadIdx.x & 31;
    int half = lane >> 4;                        // lanes 16-31 hold K+2,K+3
    int mn   = lane & 15;

    const float* arow = A + (size_t)(strip * 16 + mn) * KDIM;

    v8f acc[NT] = {};
#pragma unroll
    for (int k0 = 0; k0 < KDIM; k0 += 4) {
        int k  = k0 + 2 * half;                  // this lane-half's K base
        int kp = (k0 >> 1) + half;               // interleaved pair index
        v2f a = *(const v2f*)(arow + k);         // global_load_b64
        if constexpr (RELU) {
            a.x = fmaxf(a.x, 0.0f);
            a.y = fmaxf(a.y, 0.0f);
        }
        const float* brow = ldsB + (size_t)kp * FP * 2 + mn * 2;
#pragma unroll
        for (int nt = 0; nt < NT; ++nt) {
            v2f b = *(const v2f*)(brow + nt * 32);   // ds_load_b64, imm offset
            acc[nt] = __builtin_amdgcn_wmma_f32_16x16x4_f32(
                false, a, false, b, (short)0, acc[nt], false, false);
        }
    }

    // ---- store (masking only here, after all WMMAs) ----
#pragma unroll
    for (int nt = 0; nt < NT; ++nt) {
#pragma unroll
        for (int r = 0; r < 8; ++r) {
            int orow = strip * 16 + half * 8 + r;   // C/D layout: M split 0-7 / 8-15
            int ocol = nt * 16 + mn;
            if (FOUT == FP || ocol < FOUT)
                Out[(size_t)orow * FOUT + ocol] = acc[nt][r];
        }
    }
}

// ---------------- aggregation ----------------
// out[n,:] = dinv[n]^2 * hW[n,:] + bias[:]   (self-loop term + bias)
template <int CH>
__global__ void agg_init_kernel(const float4* __restrict__ hW,
                                const float*  __restrict__ dinv,
                                const float*  __restrict__ bias,
                                float4* __restrict__ out, int total)
{
    int t = blockIdx.x * blockDim.x + threadIdx.x;
    if (t >= total) return;
    int nidx = t / CH;
    int c    = t - nidx * CH;
    float w = dinv[nidx];
    w *= w;
    float4 v = hW[t];
    int f = c * 4;
    float4 r;
    r.x = v.x * w + bias[f + 0];
    r.y = v.y * w + bias[f + 1];
    r.z = v.z * w + bias[f + 2];
    r.w = v.w * w + bias[f + 3];
    out[t] = r;
}

// out[dst[e],:] += norm[e] * hW[src[e],:]  -- L2-resident fp32 atomics
template <int CH>
__global__ void edge_scatter_kernel(const int* __restrict__ src,
                                    const int* __restrict__ dst,
                                    const float* __restrict__ norm,
                                    const float4* __restrict__ hW,
                                    float* __restrict__ out, int total)
{
    int t = blockIdx.x * blockDim.x + threadIdx.x;
    if (t >= total) return;
    int e = t / CH;
    int c = t - e * CH;
    int s = src[e];
    int d = dst[e];
    float w  = norm[e];
    float4 v = hW[(size_t)s * CH + c];
    float* o = out + (size_t)d * (CH * 4) + c * 4;
    atomicAdd(o + 0, v.x * w);
    atomicAdd(o + 1, v.y * w);
    atomicAdd(o + 2, v.z * w);
    atomicAdd(o + 3, v.w * w);
}

// ---------------- host ----------------

extern "C" void kernel_launch(void* const* d_in, const int* in_sizes, int n_in,
                              void* d_out, int out_size, void* d_ws, size_t ws_size,
                              hipStream_t stream)
{
    const float* x  = (const float*)d_in[0];
    const int*   ei = (const int*)  d_in[1];     // [2, E]: row0 = src, row1 = dst
    const float* W[4] = { (const float*)d_in[2], (const float*)d_in[4],
                          (const float*)d_in[6], (const float*)d_in[8] };
    const float* bia[4] = { (const float*)d_in[3], (const float*)d_in[5],
                            (const float*)d_in[7], (const float*)d_in[9] };
    const int* src = ei;
    const int* dst = ei + N_EDGES;

    char* ws = (char*)d_ws;
    size_t off = 0;
    auto alloc = [&](size_t bytes) {
        size_t p = off;
        off = (off + bytes + 255) & ~(size_t)255;
        return p;
    };
    float* dinv = (float*)(ws + alloc((size_t)N_NODES * 4));
    float* nrm  = (float*)(ws + alloc((size_t)N_EDGES * 4));
    float* T    = (float*)(ws + alloc((size_t)N_NODES * HDIM * 4));  // hW scratch
    float* P    = (float*)(ws + alloc((size_t)N_NODES * HDIM * 4));  // layer output

    const int TPB = 256;
    const int strips = N_NODES / 16;             // 6250, N % 16 == 0
    const int gemmBlocks = (strips + 7) / 8;     // 8 waves per block

    // gcn_norm
    init_deg_kernel<<<(N_NODES + TPB - 1) / TPB, TPB, 0, stream>>>(dinv, N_NODES);
    edge_deg_kernel<<<(N_EDGES + TPB - 1) / TPB, TPB, 0, stream>>>(dst, dinv, N_EDGES);
    dinv_kernel<<<(N_NODES + TPB - 1) / TPB, TPB, 0, stream>>>(dinv, N_NODES);
    edge_norm_kernel<<<(N_EDGES + TPB - 1) / TPB, TPB, 0, stream>>>(src, dst, dinv, nrm, N_EDGES);

    // ---- layer 0: x @ W0, no input ReLU ----
    gcn_gemm_wmma_kernel<128, false><<<gemmBlocks, TPB, 0, stream>>>(x, W[0], T, strips);
    {
        int ti = N_NODES * 32, ts = N_EDGES * 32;
        agg_init_kernel<32><<<(ti + TPB - 1) / TPB, TPB, 0, stream>>>((const float4*)T, dinv, bia[0], (float4*)P, ti);
        edge_scatter_kernel<32><<<(ts + TPB - 1) / TPB, TPB, 0, stream>>>(src, dst, nrm, (const float4*)T, P, ts);
    }
    // ---- layers 1,2: ReLU fused into A-operand load ----
    for (int l = 1; l <= 2; ++l) {
        gcn_gemm_wmma_kernel<128, true><<<gemmBlocks, TPB, 0, stream>>>(P, W[l], T, strips);
        int ti = N_NODES * 32, ts = N_EDGES * 32;
        agg_init_kernel<32><<<(ti + TPB - 1) / TPB, TPB, 0, stream>>>((const float4*)T, dinv, bia[l], (float4*)P, ti);
        edge_scatter_kernel<32><<<(ts + TPB - 1) / TPB, TPB, 0, stream>>>(src, dst, nrm, (const float4*)T, P, ts);
    }
    // ---- layer 3 -> d_out [N, 40] ----
    gcn_gemm_wmma_kernel<40, true><<<gemmBlocks, TPB, 0, stream>>>(P, W[3], T, strips);
    {
        int ti = N_NODES * 10, ts = N_EDGES * 10;
        agg_init_kernel<10><<<(ti + TPB - 1) / TPB, TPB, 0, stream>>>((const float4*)T, dinv, bia[3], (float4*)d_out, ti);
        edge_scatter_kernel<10><<<(ts + TPB - 1) / TPB, TPB, 0, stream>>>(src, dst, nrm, (const float4*)T, (float*)d_out, ts);
    }
}